// NodeRoIPool_833223655740
// MI455X (gfx1250) — compile-verified
//
#include <hip/hip_runtime.h>

typedef float v2f __attribute__((ext_vector_type(2)));
typedef float v8f __attribute__((ext_vector_type(8)));

#define CC 256
#define HH 256
#define WW 256
#define HW (HH * WW)
#define SCALE 0.25f
#define STRIDE 2

// One wave32 per (roi, point) pair.
//   lane = (h, m): m = lane & 15 (A-matrix row = channel-within-tile),
//                  h = lane >> 4 (supplies K = {2h, 2h+1} of each 4-wide WMMA).
// Per channel-tile (16 channels): 4 chained V_WMMA_F32_16X16X4_F32 with
// B = 1/16 everywhere compute the 4x4-window mean for 16 channels directly
// in the matrix pipe (no cross-lane shuffles, 2 scalar loads per lane per WMMA).
__global__ __launch_bounds__(256) void roipool_wmma_kernel(
    const float* __restrict__ feat, const float* __restrict__ rois,
    float* __restrict__ out, int npairs)
{
    const int wave = threadIdx.x >> 5;
    const int lane = threadIdx.x & 31;
    const int pair = blockIdx.x * 8 + wave;      // uniform within the wave
    if (pair >= npairs) return;

    const int roi = pair / 5;
    const int pt  = pair - roi * 5;

    // ---- point selection (matches reference: mids of rolled corners + center)
    const float* r = rois + (size_t)roi * 8;
    float xs[4], ys[4];
#pragma unroll
    for (int i = 0; i < 4; ++i) {
        xs[i] = r[2 * i]     * SCALE;
        ys[i] = r[2 * i + 1] * SCALE;
    }
    float fx, fy;
    if (pt == 4) {
        fx = (xs[0] + xs[1] + xs[2] + xs[3]) * 0.25f;
        fy = (ys[0] + ys[1] + ys[2] + ys[3]) * 0.25f;
    } else {
        const int j = (pt + 1) & 3;
        fx = (xs[pt] + xs[j]) * 0.5f;
        fy = (ys[pt] + ys[j]) * 0.5f;
    }
    const int xc = (int)fminf(fmaxf(ceilf(fx), (float)STRIDE), (float)(WW - STRIDE));
    const int yc = (int)fminf(fmaxf(ceilf(fy), (float)STRIDE), (float)(HH - STRIDE));
    const int x0 = xc - STRIDE;
    const int y0 = yc - STRIDE;

    const int m = lane & 15;
    const int h = lane >> 4;

    // lane's gather base: channel m, window row y0, columns x0 + 2h + {0,1}
    const float* base = feat + (size_t)m * HW + (size_t)y0 * WW + (x0 + 2 * h);
    float* obase = out + (size_t)pair * CC;

    const v2f bmat = {0.0625f, 0.0625f};  // B = ones/16 -> D = window mean

    for (int t = 0; t < 16; ++t) {
        const float* p = base + (size_t)t * (16 * HW);
        if (t < 15) __builtin_prefetch(p + 16 * HW, 0, 1);  // next channel tile

        v8f acc = {};
#pragma unroll
        for (int g = 0; g < 4; ++g) {        // K-group g = window row y0+g
            v2f a;
            a.x = p[g * WW];
            a.y = p[g * WW + 1];
            acc = __builtin_amdgcn_wmma_f32_16x16x4_f32(
                false, a, false, bmat, (short)0, acc, false, false);
        }

        // D layout: lane h*16+m, VGPR r holds row (h*8 + r); columns replicated.
        // Lanes with m < 8 store channels t*16 + h*8 + m  (32B contiguous/half).
        if (m < 8) {
            float v = acc[0];
            v = (m == 1) ? acc[1] : v;
            v = (m == 2) ? acc[2] : v;
            v = (m == 3) ? acc[3] : v;
            v = (m == 4) ? acc[4] : v;
            v = (m == 5) ? acc[5] : v;
            v = (m == 6) ? acc[6] : v;
            v = (m == 7) ? acc[7] : v;
            obase[t * 16 + h * 8 + m] = v;
        }
    }
}

extern "C" void kernel_launch(void* const* d_in, const int* in_sizes, int n_in,
                              void* d_out, int out_size, void* d_ws, size_t ws_size,
                              hipStream_t stream) {
    const float* feat = (const float*)d_in[0];
    const float* rois = (const float*)d_in[1];
    float* out = (float*)d_out;

    const int nrois  = in_sizes[1] / 8;
    const int npairs = nrois * 5;            // 100000 for the reference shapes

    const int waves_per_block = 8;           // 256 threads = 8 wave32
    dim3 block(256);
    dim3 grid((npairs + waves_per_block - 1) / waves_per_block);
    roipool_wmma_kernel<<<grid, block, 0, stream>>>(feat, rois, out, npairs);
}